// SAGEConv_1580547970266
// MI455X (gfx1250) — compile-verified
//
#include <hip/hip_runtime.h>

typedef __attribute__((ext_vector_type(2))) float v2f;
typedef __attribute__((ext_vector_type(8))) float v8f;

#define IN_DIM  256
#define P_DIM   128
#define OUT_DIM 256

// Native fp32 atomic add into global memory (L2-resident agg buffer).
// Forced via inline asm: clang-22 dropped __builtin_amdgcn_global_atomic_fadd_f32,
// and we must NOT fall back to a CAS loop for the 205M contended adds.
// No-return form -> tracked on STOREcnt, no VGPR writeback, fire-and-forget.
__device__ __forceinline__ void atomic_add_f32(float* p, float v) {
    asm volatile("global_atomic_add_f32 %0, %1, off"
                 :
                 : "v"(p), "v"(v)
                 : "memory");
}

// ---------------------------------------------------------------------------
// Phase 1: h = relu(x @ Wp^T + bp)   [N,128]
// One block = 16 rows of x staged in LDS; 8 waves, each owns one 16-col tile
// of P=128. fp32 WMMA 16x16x4, K swept over IN=256.
// ---------------------------------------------------------------------------
__global__ __launch_bounds__(256) void pool_gemm_relu(
        const float* __restrict__ x, const float* __restrict__ Wp,
        const float* __restrict__ bp, float* __restrict__ h) {
    __shared__ float xs[16 * IN_DIM];           // 16 KB

    const int r0 = blockIdx.x * 16;
    // 16 consecutive rows of x are contiguous: linear float4 copy to LDS.
    const float4* s4 = reinterpret_cast<const float4*>(x + (size_t)r0 * IN_DIM);
    float4* d4 = reinterpret_cast<float4*>(xs);
    for (int i = threadIdx.x; i < (16 * IN_DIM) / 4; i += blockDim.x) d4[i] = s4[i];
    __syncthreads();

    const int wave = threadIdx.x >> 5;
    const int lane = threadIdx.x & 31;
    const int mn   = lane & 15;     // A row / B,C col handled by this lane
    const int kh   = lane >> 4;     // K-half select
    const int c0   = wave * 16;     // output column tile base

    const float* Arow = xs + mn * IN_DIM;                     // LDS
    const float* Brow = Wp + (size_t)(c0 + mn) * IN_DIM;      // Wp row = Wp^T col
    __builtin_prefetch(Brow, 0, 0);

    v8f acc = {};
#pragma unroll 4
    for (int k = 0; k < IN_DIM; k += 4) {
        v2f a = *reinterpret_cast<const v2f*>(Arow + k + 2 * kh);
        v2f b = *reinterpret_cast<const v2f*>(Brow + k + 2 * kh);
        acc = __builtin_amdgcn_wmma_f32_16x16x4_f32(
            false, a, false, b, (short)0, acc, false, false);
    }

    const float bias = bp[c0 + mn];
    float* hb = h + (size_t)r0 * P_DIM + c0 + mn;
#pragma unroll
    for (int j = 0; j < 8; ++j) {
        const int row = j + 8 * kh;
        float v = acc[j] + bias;
        hb[row * P_DIM] = v > 0.f ? v : 0.f;
    }
}

// ---------------------------------------------------------------------------
// Phase 2: zero the aggregation buffer (ws is poisoned; must re-zero per call).
// ---------------------------------------------------------------------------
__global__ void zero_agg(float* __restrict__ p, long n) {
    long i = (long)blockIdx.x * blockDim.x + threadIdx.x;
    const long stride = (long)gridDim.x * blockDim.x;
    for (; i < n; i += stride) p[i] = 0.f;
}

// ---------------------------------------------------------------------------
// Phase 3: agg[dst] += edge_vals * h[src]^2. One wave per edge; each lane
// handles 4 of the 128 features (float4 gather, 4 native f32 atomics).
// h and agg (51 MB each) are L2-resident -> this is L2-atomic bound.
// ---------------------------------------------------------------------------
__global__ __launch_bounds__(256) void edge_scatter(
        const float* __restrict__ h, const float* __restrict__ edge_vals,
        const int* __restrict__ src, const int* __restrict__ dst,
        float* __restrict__ agg, int E) {
    const int e = blockIdx.x * 8 + (threadIdx.x >> 5);
    if (e >= E) return;
    const int lane = threadIdx.x & 31;

    const int   s  = src[e];
    const int   d  = dst[e];
    const float ev = edge_vals[e];

    const float4 hv =
        reinterpret_cast<const float4*>(h + (size_t)s * P_DIM)[lane];
    float* ab = agg + (size_t)d * P_DIM + lane * 4;
    atomic_add_f32(ab + 0, ev * hv.x * hv.x);   // MU = 2 -> h^2
    atomic_add_f32(ab + 1, ev * hv.y * hv.y);
    atomic_add_f32(ab + 2, ev * hv.z * hv.z);
    atomic_add_f32(ab + 3, ev * hv.w * hv.w);
}

// ---------------------------------------------------------------------------
// Phase 4: out = h @ W1^T + sqrt(agg) @ W2^T + (b1 + b2).   [N,256]
// One block = 16 rows; 16 waves, each owns one 16-col tile of OUT=256.
// Two fp32 WMMA K-sweeps (K=128 each) into one accumulator.
// ---------------------------------------------------------------------------
__global__ __launch_bounds__(512) void combine_gemm(
        const float* __restrict__ h, const float* __restrict__ agg,
        const float* __restrict__ W1, const float* __restrict__ b1,
        const float* __restrict__ W2, const float* __restrict__ b2,
        float* __restrict__ out) {
    __shared__ float hs[16 * P_DIM];            // 8 KB
    __shared__ float gs[16 * P_DIM];            // 8 KB (sqrt(agg))

    const int r0 = blockIdx.x * 16;
    const float4* h4 = reinterpret_cast<const float4*>(h + (size_t)r0 * P_DIM);
    const float4* a4 = reinterpret_cast<const float4*>(agg + (size_t)r0 * P_DIM);
    float4* hs4 = reinterpret_cast<float4*>(hs);
    float4* gs4 = reinterpret_cast<float4*>(gs);
    for (int i = threadIdx.x; i < (16 * P_DIM) / 4; i += blockDim.x) {
        hs4[i] = h4[i];
        float4 v = a4[i];                       // agg >= 0 by construction
        v.x = sqrtf(v.x); v.y = sqrtf(v.y); v.z = sqrtf(v.z); v.w = sqrtf(v.w);
        gs4[i] = v;                             // MU = 2 -> agg^(1/2)
    }
    __syncthreads();

    const int wave = threadIdx.x >> 5;
    const int lane = threadIdx.x & 31;
    const int mn   = lane & 15;
    const int kh   = lane >> 4;
    const int c0   = wave * 16;

    const float* A1 = hs + mn * P_DIM;
    const float* A2 = gs + mn * P_DIM;
    const float* B1 = W1 + (size_t)(c0 + mn) * P_DIM;
    const float* B2 = W2 + (size_t)(c0 + mn) * P_DIM;
    __builtin_prefetch(B1, 0, 0);
    __builtin_prefetch(B2, 0, 0);

    v8f acc = {};
#pragma unroll 4
    for (int k = 0; k < P_DIM; k += 4) {
        v2f a = *reinterpret_cast<const v2f*>(A1 + k + 2 * kh);
        v2f b = *reinterpret_cast<const v2f*>(B1 + k + 2 * kh);
        acc = __builtin_amdgcn_wmma_f32_16x16x4_f32(
            false, a, false, b, (short)0, acc, false, false);
    }
#pragma unroll 4
    for (int k = 0; k < P_DIM; k += 4) {
        v2f a = *reinterpret_cast<const v2f*>(A2 + k + 2 * kh);
        v2f b = *reinterpret_cast<const v2f*>(B2 + k + 2 * kh);
        acc = __builtin_amdgcn_wmma_f32_16x16x4_f32(
            false, a, false, b, (short)0, acc, false, false);
    }

    const float bias = b1[c0 + mn] + b2[c0 + mn];
    float* ob = out + (size_t)r0 * OUT_DIM + c0 + mn;
#pragma unroll
    for (int j = 0; j < 8; ++j) {
        ob[(j + 8 * kh) * OUT_DIM] = acc[j] + bias;
    }
}

// ---------------------------------------------------------------------------
extern "C" void kernel_launch(void* const* d_in, const int* in_sizes, int n_in,
                              void* d_out, int out_size, void* d_ws, size_t ws_size,
                              hipStream_t stream) {
    const float* x         = (const float*)d_in[0];
    const float* edge_vals = (const float*)d_in[1];
    const float* Wp        = (const float*)d_in[2];
    const float* bp        = (const float*)d_in[3];
    const float* W1        = (const float*)d_in[4];
    const float* b1        = (const float*)d_in[5];
    const float* W2        = (const float*)d_in[6];
    const float* b2        = (const float*)d_in[7];
    const int*   src       = (const int*)d_in[8];
    const int*   dst       = (const int*)d_in[9];
    float*       out       = (float*)d_out;

    const int N = in_sizes[0] / IN_DIM;     // 100000
    const int E = in_sizes[1];              // 1600000

    // Workspace: h [N*P] then agg [N*P]  (2 * 51.2 MB)
    float* h   = (float*)d_ws;
    float* agg = h + (size_t)N * P_DIM;

    pool_gemm_relu<<<N / 16, 256, 0, stream>>>(x, Wp, bp, h);
    zero_agg<<<2048, 256, 0, stream>>>(agg, (long)N * P_DIM);
    edge_scatter<<<(E + 7) / 8, 256, 0, stream>>>(h, edge_vals, src, dst, agg, E);
    combine_gemm<<<N / 16, 512, 0, stream>>>(h, agg, W1, b1, W2, b2, out);
}